// CenterLoss_35759897706846
// MI455X (gfx1250) — compile-verified
//
#include <hip/hip_runtime.h>

typedef __attribute__((ext_vector_type(2))) float v2f;
typedef __attribute__((ext_vector_type(8))) float v8f;

#define FEAT   128
#define NCLS   1000
#define BATCH  32768
#define WAVES_PER_BLOCK 8          // 256 threads = 8 wave32
#define ROWS_PER_WAVE   16
#define NBLOCKS (BATCH / (ROWS_PER_WAVE * WAVES_PER_BLOCK))   // 256

__global__ void CenterLoss_zero_ws(double* p) { p[0] = 0.0; }

__global__ __launch_bounds__(256) void CenterLoss_main(
    const float* __restrict__ y,
    const int*   __restrict__ labels,
    const float* __restrict__ centers,
    double*      __restrict__ partial)
{
    const int lane = threadIdx.x & 31;
    const int wave = threadIdx.x >> 5;
    const int tile = blockIdx.x * WAVES_PER_BLOCK + wave;   // 16 rows per tile
    const int hi   = lane >> 4;        // which half of the wave (K pairs 0/1 vs 2/3)
    const int m    = lane & 15;        // row (A) / col (B) index within tile

    const int row = tile * ROWS_PER_WAVE + m;
    const int lab = labels[row];

    const float* __restrict__ yrow = y       + (size_t)row * FEAT;
    const float* __restrict__ crow = centers + (size_t)lab * FEAT;
    const int coff = 2 * hi;           // K offset handled by this half-wave

    v8f acc = {};                      // 16x16 f32 accumulator (Y * gathered-C^T)
    float ysq = 0.0f, csq = 0.0f;

    #pragma unroll 8
    for (int kk = 0; kk < FEAT / 4; ++kk) {
        const int k0 = 4 * kk + coff;
        // A: 16x4 f32 tile of y   (vgpr0 = K even, vgpr1 = K odd per ISA layout)
        v2f a = *(const v2f*)(yrow + k0);
        // B: 4x16 f32 tile of gathered centers (mirrored layout)
        v2f b = *(const v2f*)(crow + k0);
        ysq += a.x * a.x + a.y * a.y;
        csq += b.x * b.x + b.y * b.y;
        // D = A*B + C  (V_WMMA_F32_16X16X4_F32), f32 in/out, RNE — matches reference math
        acc = __builtin_amdgcn_wmma_f32_16x16x4_f32(
                  false, a, false, b, (short)0, acc, false, false);
    }

    // Each lane summed half the columns of its row; lanes m and m+16 complete it.
    ysq += __shfl_xor(ysq, 16, 32);
    csq += __shfl_xor(csq, 16, 32);

    // Diagonal C[m][m]: held by lane m+16*(m>=8) in acc vgpr (m%8).
    const int v = m - 8 * hi;                     // valid iff 0 <= v < 8
    float t01 = (v & 1) ? acc[1] : acc[0];
    float t23 = (v & 1) ? acc[3] : acc[2];
    float t45 = (v & 1) ? acc[5] : acc[4];
    float t67 = (v & 1) ? acc[7] : acc[6];
    float t03 = (v & 2) ? t23 : t01;
    float t47 = (v & 2) ? t67 : t45;
    float dot = (v & 4) ? t47 : t03;

    float contrib = 0.0f;
    if ((unsigned)v < 8u) {
        float dist = ysq + csq - 2.0f * dot;      // expanded form, as in reference
        dist = fminf(fmaxf(dist, 1e-12f), 1e12f); // clip
        contrib = dist;
    }

    // Wave32 reduction
    #pragma unroll
    for (int off = 16; off > 0; off >>= 1)
        contrib += __shfl_xor(contrib, off, 32);

    __shared__ float wsum[WAVES_PER_BLOCK];
    if (lane == 0) wsum[wave] = contrib;
    __syncthreads();
    if (threadIdx.x == 0) {
        float s = 0.0f;
        #pragma unroll
        for (int i = 0; i < WAVES_PER_BLOCK; ++i) s += wsum[i];
        atomicAdd(partial, (double)s);            // global_atomic_add_f64
    }
}

__global__ void CenterLoss_finalize(const double* __restrict__ partial,
                                    const float*  __restrict__ lw,
                                    float*        __restrict__ out)
{
    // masked-out entries: B*(C-1) zeros each clipped up to 1e-12
    double loss = (partial[0] / (double)BATCH + (double)(NCLS - 1) * 1e-12)
                  * (double)lw[0];
    out[0] = (float)loss;
}

extern "C" void kernel_launch(void* const* d_in, const int* in_sizes, int n_in,
                              void* d_out, int out_size, void* d_ws, size_t ws_size,
                              hipStream_t stream) {
    const float* y       = (const float*)d_in[0];
    const int*   labels  = (const int*)  d_in[1];
    const float* centers = (const float*)d_in[2];
    const float* lw      = (const float*)d_in[3];
    double* partial = (double*)d_ws;

    CenterLoss_zero_ws<<<1, 1, 0, stream>>>(partial);
    CenterLoss_main<<<NBLOCKS, WAVES_PER_BLOCK * 32, 0, stream>>>(y, labels, centers, partial);
    CenterLoss_finalize<<<1, 1, 0, stream>>>(partial, lw, (float*)d_out);
}